// GraphConvolution_2362232012852
// MI455X (gfx1250) — compile-verified
//
#include <hip/hip_runtime.h>

typedef __attribute__((ext_vector_type(2))) float v2f;
typedef __attribute__((ext_vector_type(4))) float v4f;
typedef __attribute__((ext_vector_type(8))) float v8f;

#define N_NODES 50000
#define N_EDGES 800000
#define D 64                     // D_IN == D_OUT == 64
#define M_TILES (N_NODES / 16)   // 3125 (exact)
#define STR2 80                  // padded float2 row stride: 640B -> +32 bank shift per row

// ---------------------------------------------------------------------------
// Kernel 1: support[50000x64] = X[50000x64] @ W[64x64]
// One wave32 computes a 16(M) x 64(N) strip via 4 accumulators, looping K in
// steps of 4 with v_wmma_f32_16x16x4_f32 (full fp32 precision).
// W is staged in LDS K-pair-packed: lWp[k/2][n] = {W[k][n], W[k+1][n]}, so
// each B fragment is a single bank-conflict-free ds_load_b64 directly into an
// even-aligned VGPR pair (no repacking movs before the WMMA).
// ---------------------------------------------------------------------------
__global__ __launch_bounds__(256) void gcn_gemm_xw(const float* __restrict__ X,
                                                   const float* __restrict__ W,
                                                   float* __restrict__ support) {
    __shared__ v2f lWp[(D / 2) * STR2];   // 32 rows x 80 float2 = 20 KB

    // Cooperative K-pair-packed fill (one-time, coalesced b32 reads of W).
    for (int i = threadIdx.x; i < (D / 2) * D; i += 256) {
        int kp = i >> 6;          // k/2  : 0..31
        int n  = i & 63;          // col  : 0..63
        v2f p;
        p.x = W[(2 * kp    ) * D + n];
        p.y = W[(2 * kp + 1) * D + n];
        lWp[kp * STR2 + n] = p;
    }
    __syncthreads();

    const int lane  = threadIdx.x & 31;
    const int wid   = threadIdx.x >> 5;
    const int mtile = blockIdx.x * 8 + wid;          // wave-uniform
    if (mtile >= M_TILES) return;                    // uniform exit: EXEC stays all-1s

    const int mrow = mtile * 16 + (lane & 15);       // A-matrix row for this lane
    const int koff = (lane & 16) >> 3;               // lanes 0-15 -> K+0,1 ; 16-31 -> K+2,3
    const int n    = lane & 15;                      // column within a 16-wide N tile

    const float* xrow = X + (size_t)mrow * D;

    v8f c0 = {}, c1 = {}, c2 = {}, c3 = {};

    #pragma unroll
    for (int kk = 0; kk < D; kk += 4) {
        // A fragment: consecutive K pair -> single 8-byte load
        v2f a = *(const v2f*)(xrow + kk + koff);

        // B fragments: one ds_load_b64 each; lower/upper half-waves read
        // adjacent kpair rows, padded apart by 32 banks -> conflict-free.
        const v2f* brow = lWp + ((kk + koff) >> 1) * STR2 + n;
        v2f b0 = brow[0];
        v2f b1 = brow[16];
        v2f b2 = brow[32];
        v2f b3 = brow[48];

        c0 = __builtin_amdgcn_wmma_f32_16x16x4_f32(false, a, false, b0, (short)0, c0, false, false);
        c1 = __builtin_amdgcn_wmma_f32_16x16x4_f32(false, a, false, b1, (short)0, c1, false, false);
        c2 = __builtin_amdgcn_wmma_f32_16x16x4_f32(false, a, false, b2, (short)0, c2, false, false);
        c3 = __builtin_amdgcn_wmma_f32_16x16x4_f32(false, a, false, b3, (short)0, c3, false, false);
    }

    // C/D layout: VGPR r -> row r (lanes 0-15) / row r+8 (lanes 16-31), col = lane&15
    const int rbase = mtile * 16 + ((lane & 16) >> 1);   // +8 for upper half-wave
    #pragma unroll
    for (int r = 0; r < 8; ++r) {
        float* o = support + (size_t)(rbase + r) * D + n;
        o[0]  = c0[r];
        o[16] = c1[r];
        o[32] = c2[r];
        o[48] = c3[r];
    }
}

// ---------------------------------------------------------------------------
// Kernel 2: out[n][f] = b[f]  (bias broadcast; scatter then accumulates on top)
// ---------------------------------------------------------------------------
__global__ void gcn_init_bias(const float* __restrict__ b, float* __restrict__ out) {
    int i = blockIdx.x * blockDim.x + threadIdx.x;      // float4 index
    if (i < (N_NODES * D) / 4) {
        ((v4f*)out)[i] = ((const v4f*)b)[i & (D / 4 - 1)];
    }
}

// ---------------------------------------------------------------------------
// Kernel 3: out[dst[e]] += w[e] * support[src[e]]   (51.2M f32 atomics, L2-resident)
// 16 lanes per edge, float4 per lane -> one full 64-float row per half-wave.
// ---------------------------------------------------------------------------
__global__ void gcn_scatter_add(const float* __restrict__ support,
                                const int* __restrict__ esrc,
                                const int* __restrict__ edst,
                                const float* __restrict__ ew,
                                float* __restrict__ out) {
    int t = blockIdx.x * blockDim.x + threadIdx.x;
    int e = t >> 4;
    if (e >= N_EDGES) return;
    int f = (t & 15) << 2;

    int s = esrc[e];
    int d = edst[e];
    float w = ew[e];

    v4f v = *(const v4f*)(support + (size_t)s * D + f);
    float* o = out + (size_t)d * D + f;
    atomicAdd(o + 0, w * v.x);
    atomicAdd(o + 1, w * v.y);
    atomicAdd(o + 2, w * v.z);
    atomicAdd(o + 3, w * v.w);
}

// ---------------------------------------------------------------------------
extern "C" void kernel_launch(void* const* d_in, const int* in_sizes, int n_in,
                              void* d_out, int out_size, void* d_ws, size_t ws_size,
                              hipStream_t stream) {
    const float* X  = (const float*)d_in[0];
    const int* esrc = (const int*)d_in[1];
    const int* edst = (const int*)d_in[2];
    const float* ew = (const float*)d_in[3];
    const float* W  = (const float*)d_in[4];
    const float* b  = (const float*)d_in[5];
    float* out      = (float*)d_out;
    float* support  = (float*)d_ws;    // 50000*64*4 = 12.8 MB scratch

    // 1) support = X @ W   (8 M-tiles per 256-thread block)
    gcn_gemm_xw<<<(M_TILES + 7) / 8, 256, 0, stream>>>(X, W, support);

    // 2) out = broadcast(b)
    gcn_init_bias<<<((N_NODES * D / 4) + 255) / 256, 256, 0, stream>>>(b, out);

    // 3) out[dst] += w * support[src]
    int scatter_blocks = (N_EDGES * 16) / 256;   // 50000 blocks exactly
    gcn_scatter_add<<<scatter_blocks, 256, 0, stream>>>(support, esrc, edst, ew, out);
}